// GenuineFractalAnalyzer_75161927680570
// MI455X (gfx1250) — compile-verified
//
#include <hip/hip_runtime.h>
#include <math.h>

typedef float v2f __attribute__((ext_vector_type(2)));
typedef float v8f __attribute__((ext_vector_type(8)));

#define WS_SUMSQ 0
#define WS_MINK  1
#define WS_MAXK  2
#define WS_OCC   4     // 256 floats
#define WS_C0    260   // start per box, 256 floats
#define WS_C1    516   // end  per box, 256 floats
#define TILE     2048

__device__ __forceinline__ unsigned okey(float f) {
  unsigned b = __float_as_uint(f);
  return (b & 0x80000000u) ? ~b : (b | 0x80000000u);
}
__device__ __forceinline__ float odec(unsigned k) {
  return __uint_as_float((k & 0x80000000u) ? (k ^ 0x80000000u) : ~k);
}
// sigmoid(a)*sigmoid(b) = 1/((1+e^-a)(1+e^-b)); here a,b arrive pre-scaled
// by -log2(e) from the WMMA, so e^-a == 2^d and v_exp_f32 needs no fixup.
__device__ __forceinline__ float boxp2(float d1, float d2) {
  float ea = __builtin_amdgcn_exp2f(d1);
  float eb = __builtin_amdgcn_exp2f(d2);
  return __builtin_amdgcn_rcpf((1.0f + ea) * (1.0f + eb));
}

// ---------------- init: zero accumulators, precompute box edges -----------
__global__ void k_init(float* ws) {
  int t = threadIdx.x;  // 256 threads
  if (t == 0) {
    ws[WS_SUMSQ] = 0.0f;
    ((unsigned*)ws)[WS_MINK] = 0xFFFFFFFFu;  // ordered-key min identity
    ((unsigned*)ws)[WS_MAXK] = 0x00000000u;  // ordered-key max identity
  }
  ws[WS_OCC + t] = 0.0f;
  const int nb[10] = {100, 60, 36, 22, 13, 8, 5, 3, 2, 1};
  int base = 0;
  float st = 0.0f, en = 0.0f;
  for (int i = 0; i < 10; i++) {
    if (t >= base && t < base + nb[i]) {
      int j = t - base;
      float inv = 1.0f / (float)nb[i];
      st = (float)j * inv;        // start edge
      en = (float)(j + 1) * inv;  // end edge
    }
    base += nb[i];
  }
  ws[WS_C0 + t] = st;
  ws[WS_C1 + t] = en;
}

// ---------------- pass 1: sumsq, min, max --------------------------------
__global__ void k_reduce(const float* __restrict__ x, int n, float* ws) {
  __shared__ float ssum[256];
  __shared__ float smin[256];
  __shared__ float smax[256];
  int t = threadIdx.x;
  float acc = 0.0f, mn = 3.402823466e38f, mx = -3.402823466e38f;
  for (int i = blockIdx.x * 256 + t; i < n; i += gridDim.x * 256) {
    float v = x[i];
    acc += v * v;
    mn = fminf(mn, v);
    mx = fmaxf(mx, v);
  }
  ssum[t] = acc; smin[t] = mn; smax[t] = mx;
  __syncthreads();
  for (int s = 128; s > 0; s >>= 1) {
    if (t < s) {
      ssum[t] += ssum[t + s];
      smin[t] = fminf(smin[t], smin[t + s]);
      smax[t] = fmaxf(smax[t], smax[t + s]);
    }
    __syncthreads();
  }
  if (t == 0) {
    atomicAdd(&ws[WS_SUMSQ], ssum[0]);
    atomicMin((unsigned*)ws + WS_MINK, okey(smin[0]));
    atomicMax((unsigned*)ws + WS_MAXK, okey(smax[0]));
  }
}

// ---------------- pass 2: WMMA occupancy ---------------------------------
// Block = 256 threads = 8 waves. Wave w owns boxes [32w, 32w+32) as two
// 16-box WMMA tiles. V_WMMA_F32_16X16X4_F32 generates the 16x16 tile of
// exp2-ready sigmoid arguments: d1' = -10*log2e*(x-start), d2' =
// -10*log2e*(end-x), with the log2e factor folded into the B constants.
__global__ __launch_bounds__(256) void k_occ(const float* __restrict__ x,
                                             int n, float* __restrict__ ws) {
  __shared__ float lx[TILE];
  int t = threadIdx.x;
  int lane = t & 31, w = t >> 5, l16 = lane & 15;
  bool lo = lane < 16;

  const unsigned* uw = (const unsigned*)ws;
  float mn = odec(uw[WS_MINK]);
  float mx = odec(uw[WS_MAXK]);
  float inv = 1.0f / (mx - mn + 1e-8f);

  const float* c0 = ws + WS_C0;
  const float* c1 = ws + WS_C1;
  int nA = (w << 5) + l16;
  int nB = nA + 16;

  const float L10 = 14.4269504088896340736f;  // 10*log2(e)
  // B rows K=2,3 (upper lanes) are zero so A's upper-lane garbage cancels.
  v2f B1A, B2A, B1B, B2B;
  B1A[0] = lo ? -L10 : 0.0f;  B1A[1] = lo ?  (L10 * c0[nA]) : 0.0f;
  B2A[0] = lo ?  L10 : 0.0f;  B2A[1] = lo ? -(L10 * c1[nA]) : 0.0f;
  B1B[0] = lo ? -L10 : 0.0f;  B1B[1] = lo ?  (L10 * c0[nB]) : 0.0f;
  B2B[0] = lo ?  L10 : 0.0f;  B2B[1] = lo ? -(L10 * c1[nB]) : 0.0f;

  float accA = 0.0f, accB = 0.0f;
  int numTiles = (n + TILE - 1) / TILE;
  int stride = gridDim.x * TILE;

  for (int tb = blockIdx.x; tb < numTiles; tb += gridDim.x) {
    int base = tb * TILE;
    __syncthreads();
    for (int i = t; i < TILE; i += 256) {
      int g = base + i;
      // pad with x=1e4: d2' -> +1.4e5, 2^that = inf, rcp(inf) = 0 exactly
      lx[i] = (g < n) ? ((x[g] - mn) * inv) : 1.0e4f;
    }
    // speculative prefetch of next grid-stride tile (dropped if OOB)
    __builtin_prefetch(x + base + stride + t * 8, 0, 1);
    __syncthreads();

    for (int c = 0; c < TILE; c += 16) {
      v2f A;
      A[0] = lx[c + l16];  // K=0 (lanes 0-15) = x_m
      A[1] = 1.0f;         // K=1 = 1
      v8f z = {};
      v8f d1 = __builtin_amdgcn_wmma_f32_16x16x4_f32(false, A, false, B1A,
                                                     (short)0, z, false, false);
      v8f d2 = __builtin_amdgcn_wmma_f32_16x16x4_f32(false, A, false, B2A,
                                                     (short)0, z, false, false);
#pragma unroll
      for (int r = 0; r < 8; r++) accA += boxp2(d1[r], d2[r]);
      d1 = __builtin_amdgcn_wmma_f32_16x16x4_f32(false, A, false, B1B,
                                                 (short)0, z, false, false);
      d2 = __builtin_amdgcn_wmma_f32_16x16x4_f32(false, A, false, B2B,
                                                 (short)0, z, false, false);
#pragma unroll
      for (int r = 0; r < 8; r++) accB += boxp2(d1[r], d2[r]);
    }
  }

  // merge M=0..7 (lanes 0-15) with M=8..15 (lanes 16-31)
  accA += __shfl_xor(accA, 16, 32);
  accB += __shfl_xor(accB, 16, 32);
  if (lo) {
    atomicAdd(&ws[WS_OCC + nA], accA);
    atomicAdd(&ws[WS_OCC + nB], accB);
  }
}

// ---------------- pass 3: box counts, softmax, regression, rescale -------
__global__ void k_final(const float* __restrict__ ws,
                        const float* __restrict__ sw,
                        const float* __restrict__ fb,
                        const float* __restrict__ en,
                        float* __restrict__ out) {
  if (threadIdx.x != 0 || blockIdx.x != 0) return;
  const int nb[10] = {100, 60, 36, 22, 13, 8, 5, 3, 2, 1};
  const float* occ = ws + WS_OCC;
  float bc[10];
  int base = 0;
  for (int i = 0; i < 10; i++) {
    float s = 0.0f;
    for (int j = 0; j < nb[i]; j++) {
      float o = occ[base + j] - 0.5f;
      s += 1.0f / (1.0f + expf(-o));
    }
    bc[i] = s;
    base += nb[i];
  }
  // softmax(scale_weights)
  float wmax = sw[0];
  for (int i = 1; i < 10; i++) wmax = fmaxf(wmax, sw[i]);
  float es[10], tot = 0.0f;
  for (int i = 0; i < 10; i++) { es[i] = expf(sw[i] - wmax); tot += es[i]; }
  // least-squares slope of log(weighted count) vs log(1/scale)
  const float LN10 = 2.3025850929940457f;
  float sx = 0, sy = 0, sxy = 0, sx2 = 0;
  for (int i = 0; i < 10; i++) {
    float lsc = LN10 * (2.0f - (2.0f * (float)i) / 9.0f);  // log(1/scale_i)
    float lc = logf(bc[i] * (es[i] / tot) + 1e-8f);
    sx += lsc; sy += lc; sxy += lsc * lc; sx2 += lsc * lsc;
  }
  float nn = 10.0f;
  float slope = (nn * sxy - sx * sy) / (nn * sx2 - sx * sx + 1e-8f);
  float fd = -slope + fb[0];
  fd = fminf(fmaxf(fd, 1.0f), 2.5f);
  float Ein = sqrtf(ws[WS_SUMSQ]);
  float oe = fabsf(fd);
  out[0] = (oe > 0.0f) ? fd * en[0] * Ein / oe : fd;
}

// ---------------- launcher ------------------------------------------------
extern "C" void kernel_launch(void* const* d_in, const int* in_sizes, int n_in,
                              void* d_out, int out_size, void* d_ws, size_t ws_size,
                              hipStream_t stream) {
  const float* sig = (const float*)d_in[0];
  const float* sw  = (const float*)d_in[1];
  const float* fb  = (const float*)d_in[2];
  const float* en  = (const float*)d_in[3];
  float* out = (float*)d_out;
  float* ws  = (float*)d_ws;
  int n = in_sizes[0];

  k_init<<<1, 256, 0, stream>>>(ws);

  int rb = (n + 255) / 256;
  if (rb > 2048) rb = 2048;
  if (rb < 1) rb = 1;
  k_reduce<<<rb, 256, 0, stream>>>(sig, n, ws);

  int numTiles = (n + TILE - 1) / TILE;
  int ob = 512;
  if (ob > numTiles) ob = numTiles;
  if (ob < 1) ob = 1;
  k_occ<<<ob, 256, 0, stream>>>(sig, n, ws);

  k_final<<<1, 32, 0, stream>>>(ws, sw, fb, en, out);
}